// Registration_32203664785542
// MI455X (gfx1250) — compile-verified
//
#include <hip/hip_runtime.h>
#include <math.h>

#define VD 256                 // volume is 256^3 fp32 (64 MB, L2-resident)
#define VMAXF 255.0f           // dims - 1
#define TILE 16

typedef __attribute__((ext_vector_type(2))) float v2f;
typedef __attribute__((ext_vector_type(8))) float v8f;

// ---------------------------------------------------------------------------
// Setup: one wave32. Lane 0 does the scalar prologue (exp-map, 4x4 inverse,
// fold piv@T@piv_inv into X analytically); the two remaining *true* matrix
// products run on v_wmma_f32_16x16x4_f32 across the whole wave:
//   WMMA1:  Y        = X   @ Khat          (Khat = k_inv padded to 4x4)
//   WMMA2:  [P | PK] = rtm @ [X | Y]       (P = pose, PK[:3,:3] = Rw @ k_inv)
// A 16x4 layout : lane l -> M = l%16 ; VGPR v -> K = 2*(l/16)+v   (ISA 7.12.2)
// B 4x16 layout : lane l -> N = l%16 ; VGPR v -> K = 2*(l/16)+v   (same fold)
// D 16x16 layout: lane l -> N = l%16 ; VGPR r -> M = r + 8*(l/16)
// ---------------------------------------------------------------------------
__global__ __launch_bounds__(32) void Registration_setup_kernel(
    const float* __restrict__ rt_inv, const float* __restrict__ k_inv,
    const float* __restrict__ sdd, const float* __restrict__ isocenter,
    const float* __restrict__ origin, const float* __restrict__ spacing,
    const float* __restrict__ rot, const float* __restrict__ xyz,
    float* __restrict__ ws) {
  __shared__ float sX[4][4];   // X   = piv @ T @ piv_inv
  __shared__ float sK[4][4];   // Khat = [k_inv 0; 0 0]
  __shared__ float sR[4][4];   // rtm
  __shared__ float sY[4][4];   // X @ Khat
  __shared__ float sP[4][8];   // [pose | pose @ Khat]

  const int l = threadIdx.x;   // all 32 lanes stay active (WMMA needs EXEC=~0)

  if (l == 0) {
    // --- so3_exp_map ---
    const float w0 = rot[0], w1 = rot[1], w2 = rot[2];
    const float th2 = w0 * w0 + w1 * w1 + w2 * w2;
    const float th = sqrtf(th2 + 1e-30f);
    const float a = (th2 > 1e-8f) ? (sinf(th) / th) : (1.0f - th2 * (1.0f / 6.0f));
    const float b = (th2 > 1e-8f) ? ((1.0f - cosf(th)) / th2)
                                  : (0.5f - th2 * (1.0f / 24.0f));
    const float K[3][3] = {{0.f, -w2, w1}, {w2, 0.f, -w0}, {-w1, w0, 0.f}};
    float K2[3][3];
    for (int i = 0; i < 3; ++i)
      for (int j = 0; j < 3; ++j) {
        float s = 0.f;
        for (int k = 0; k < 3; ++k) s = fmaf(K[i][k], K[k][j], s);
        K2[i][j] = s;
      }
    float R[3][3];
    for (int i = 0; i < 3; ++i)
      for (int j = 0; j < 3; ++j)
        R[i][j] = ((i == j) ? 1.0f : 0.0f) + a * K[i][j] + b * K2[i][j];
    float t[3];
    for (int i = 0; i < 3; ++i)
      t[i] = R[i][0] * xyz[0] + R[i][1] * xyz[1] + R[i][2] * xyz[2];

    // --- inv(rtm) via Gauss-Jordan, then c = inv(rtm) @ [iso; 1] ---
    float A[4][8];
    for (int i = 0; i < 4; ++i)
      for (int j = 0; j < 4; ++j) {
        A[i][j] = rt_inv[i * 4 + j];
        A[i][4 + j] = (i == j) ? 1.0f : 0.0f;
      }
    for (int col = 0; col < 4; ++col) {
      int piv = col;
      for (int r = col + 1; r < 4; ++r)
        if (fabsf(A[r][col]) > fabsf(A[piv][col])) piv = r;
      if (piv != col)
        for (int j = 0; j < 8; ++j) {
          float tmp = A[col][j]; A[col][j] = A[piv][j]; A[piv][j] = tmp;
        }
      const float inv = 1.0f / A[col][col];
      for (int j = 0; j < 8; ++j) A[col][j] *= inv;
      for (int r = 0; r < 4; ++r) {
        if (r == col) continue;
        const float f = A[r][col];
        for (int j = 0; j < 8; ++j) A[r][j] = fmaf(-f, A[col][j], A[r][j]);
      }
    }
    float c[3];
    for (int i = 0; i < 3; ++i)
      c[i] = A[i][4] * isocenter[0] + A[i][5] * isocenter[1] +
             A[i][6] * isocenter[2] + A[i][7];

    // --- X = piv @ T @ piv_inv = [R | c - R c + t ; 0 0 0 1] ---
    for (int i = 0; i < 4; ++i)
      for (int j = 0; j < 4; ++j) {
        sX[i][j] = 0.f; sK[i][j] = 0.f;
        sR[i][j] = rt_inv[i * 4 + j];
      }
    for (int i = 0; i < 3; ++i) {
      float rc = 0.f;
      for (int k = 0; k < 3; ++k) { sX[i][k] = R[i][k]; rc = fmaf(R[i][k], c[k], rc); }
      sX[i][3] = c[i] - rc + t[i];
      for (int j = 0; j < 3; ++j) sK[i][j] = k_inv[i * 3 + j];
    }
    sX[3][3] = 1.0f;
  }
  __syncthreads();

  const int MN = l & 15;          // A: row M ; B: col N
  const int kb = (l >> 4) * 2;    // K base for this half-wave

  // ---- WMMA1: Y = X @ Khat ----
  v2f a1, b1;
  a1.x = (MN < 4) ? sX[MN][kb]     : 0.f;
  a1.y = (MN < 4) ? sX[MN][kb + 1] : 0.f;
  b1.x = (MN < 4) ? sK[kb][MN]     : 0.f;
  b1.y = (MN < 4) ? sK[kb + 1][MN] : 0.f;
  v8f cz = {};
  v8f d1 = __builtin_amdgcn_wmma_f32_16x16x4_f32(
      false, a1, false, b1, (short)0, cz, false, false);
  if (l < 4) {                    // lanes 0..3 hold column N=l, rows in d1[0..3]
    sY[0][l] = d1[0]; sY[1][l] = d1[1]; sY[2][l] = d1[2]; sY[3][l] = d1[3];
  }
  __syncthreads();

  // ---- WMMA2: [pose | pose@Khat] = rtm @ [X | Y] ----
  v2f a2, b2;
  a2.x = (MN < 4) ? sR[MN][kb]     : 0.f;
  a2.y = (MN < 4) ? sR[MN][kb + 1] : 0.f;
  if (MN < 4)      { b2.x = sX[kb][MN];     b2.y = sX[kb + 1][MN]; }
  else if (MN < 7) { b2.x = sY[kb][MN - 4]; b2.y = sY[kb + 1][MN - 4]; }
  else             { b2.x = 0.f;            b2.y = 0.f; }
  v8f d2 = __builtin_amdgcn_wmma_f32_16x16x4_f32(
      false, a2, false, b2, (short)0, cz, false, false);
  if (l < 8) {
    sP[0][l] = d2[0]; sP[1][l] = d2[1]; sP[2][l] = d2[2]; sP[3][l] = d2[3];
  }
  __syncthreads();

  if (l == 0) {
    const float s0 = sdd[0];
    for (int i = 0; i < 3; ++i)
      for (int j = 0; j < 3; ++j)
        ws[i * 3 + j] = sP[i][4 + j] * s0;        // M = Rw @ k_inv * sdd
    for (int i = 0; i < 3; ++i) {
      ws[9 + i] = (sP[i][3] - origin[i]) / spacing[i];  // src in voxel space
      ws[12 + i] = 1.0f / spacing[i];
    }
  }
}

// ---------------- main: one ray per lane, 16x16 pixel tiles ----------------
__global__ __launch_bounds__(256) void Registration_drr_kernel(
    const float* __restrict__ vol, const float* __restrict__ ws,
    const int* __restrict__ hp, const int* __restrict__ wp,
    const int* __restrict__ np, float* __restrict__ out) {
  __shared__ float sws[16];
  const int ltid = threadIdx.y * TILE + threadIdx.x;
  if (ltid < 15) sws[ltid] = ws[ltid];
  __syncthreads();

  const int W = *wp, H = *hp, N = *np;
  const int u = blockIdx.x * TILE + threadIdx.x;
  const int v = blockIdx.y * TILE + threadIdx.y;
  if (u >= W || v >= H) return;

  const float pu = (float)u + 0.5f;
  const float pv = (float)v + 0.5f;

  // world-space ray direction: dirs = M @ [pu, pv, 1]
  const float dx = fmaf(sws[0], pu, fmaf(sws[1], pv, sws[2]));
  const float dy = fmaf(sws[3], pu, fmaf(sws[4], pv, sws[5]));
  const float dz = fmaf(sws[6], pu, fmaf(sws[7], pv, sws[8]));

  const float invN = 1.0f / (float)N;
  const float seglen = sqrtf(fmaf(dx, dx, fmaf(dy, dy, dz * dz))) * invN;

  // voxel-space source and step direction
  const float sx = sws[9], sy = sws[10], sz = sws[11];
  const float dvx = dx * sws[12], dvy = dy * sws[13], dvz = dz * sws[14];

  // slab test: conservative step range where the ray can be inside [0,255]^3
  float t0 = 0.0f, t1 = 1.0f;
  if (dvx != 0.0f) {
    const float ra = (0.0f - sx) / dvx, rb = (VMAXF - sx) / dvx;
    t0 = fmaxf(t0, fminf(ra, rb)); t1 = fminf(t1, fmaxf(ra, rb));
  } else if (sx < 0.0f || sx > VMAXF) t1 = -1.0f;
  if (dvy != 0.0f) {
    const float ra = (0.0f - sy) / dvy, rb = (VMAXF - sy) / dvy;
    t0 = fmaxf(t0, fminf(ra, rb)); t1 = fminf(t1, fmaxf(ra, rb));
  } else if (sy < 0.0f || sy > VMAXF) t1 = -1.0f;
  if (dvz != 0.0f) {
    const float ra = (0.0f - sz) / dvz, rb = (VMAXF - sz) / dvz;
    t0 = fmaxf(t0, fminf(ra, rb)); t1 = fminf(t1, fmaxf(ra, rb));
  } else if (sz < 0.0f || sz > VMAXF) t1 = -1.0f;

  int iStart = (int)floorf(t0 * (float)N - 0.5f) - 1;   // +-1 margin vs rounding
  int iEnd   = (int)ceilf (t1 * (float)N - 0.5f) + 1;
  iStart = max(iStart, 0);
  iEnd   = min(iEnd, N - 1);

  float acc = 0.0f;
#pragma unroll 2
  for (int i = iStart; i <= iEnd; ++i) {
    const float frac = ((float)i + 0.5f) * invN;
    const float px = fmaf(frac, dvx, sx);
    const float py = fmaf(frac, dvy, sy);
    const float pz = fmaf(frac, dvz, sz);

    // exact reference inside-check retained (slab range is conservative)
    const bool inside = (px >= 0.0f) & (px <= VMAXF) &
                        (py >= 0.0f) & (py <= VMAXF) &
                        (pz >= 0.0f) & (pz <= VMAXF);
    if (inside) {
      const float fx0 = floorf(px), fy0 = floorf(py), fz0 = floorf(pz);
      const float fx = px - fx0, fy = py - fy0, fz = pz - fz0;
      const int ix0 = (int)fx0, iy0 = (int)fy0, iz0 = (int)fz0;
      const int ix1 = min(ix0 + 1, VD - 1);
      const int iy1 = min(iy0 + 1, VD - 1);
      const int iz1 = min(iz0 + 1, VD - 1);

      const int x0 = ix0 << 16, x1 = ix1 << 16;   // x * 256 * 256
      const int y0 = iy0 << 8,  y1 = iy1 << 8;    // y * 256

      // issue all 8 gathers back-to-back for max loads in flight
      const float c000 = vol[x0 + y0 + iz0];
      const float c001 = vol[x0 + y0 + iz1];
      const float c010 = vol[x0 + y1 + iz0];
      const float c011 = vol[x0 + y1 + iz1];
      const float c100 = vol[x1 + y0 + iz0];
      const float c101 = vol[x1 + y0 + iz1];
      const float c110 = vol[x1 + y1 + iz0];
      const float c111 = vol[x1 + y1 + iz1];

      const float gx = 1.0f - fx, gy = 1.0f - fy, gz = 1.0f - fz;
      const float gygz = gy * gz, fygz = fy * gz;
      const float gyfz = gy * fz, fyfz = fy * fz;

      float val = c000 * (gx * gygz);
      val = fmaf(c100, fx * gygz, val);
      val = fmaf(c010, gx * fygz, val);
      val = fmaf(c001, gx * gyfz, val);
      val = fmaf(c110, fx * fygz, val);
      val = fmaf(c101, fx * gyfz, val);
      val = fmaf(c011, gx * fyfz, val);
      val = fmaf(c111, fx * fyfz, val);
      acc += val;
    }
  }
  out[v * W + u] = acc * seglen;
}

// ---------------------------------------------------------------------------
extern "C" void kernel_launch(void* const* d_in, const int* in_sizes, int n_in,
                              void* d_out, int out_size, void* d_ws, size_t ws_size,
                              hipStream_t stream) {
  const float* volume    = (const float*)d_in[0];
  const float* rt_inv    = (const float*)d_in[1];
  const float* k_inv     = (const float*)d_in[2];
  const float* sdd       = (const float*)d_in[3];
  const float* isocenter = (const float*)d_in[4];
  const float* origin    = (const float*)d_in[5];
  const float* spacing   = (const float*)d_in[6];
  const float* rot       = (const float*)d_in[7];
  const float* xyz       = (const float*)d_in[8];
  const int*   height    = (const int*)d_in[9];
  const int*   width     = (const int*)d_in[10];
  const int*   n_steps   = (const int*)d_in[11];
  float* out = (float*)d_out;
  float* ws  = (float*)d_ws;

  Registration_setup_kernel<<<1, 32, 0, stream>>>(
      rt_inv, k_inv, sdd, isocenter, origin, spacing, rot, xyz, ws);

  // 512x512 image, 16x16 pixel tiles (8 wave32 per block) for ray/L2 locality
  dim3 block(TILE, TILE);
  dim3 grid((512 + TILE - 1) / TILE, (512 + TILE - 1) / TILE);
  Registration_drr_kernel<<<grid, block, 0, stream>>>(
      volume, ws, height, width, n_steps, out);
}